// Decoder_47863115546709
// MI455X (gfx1250) — compile-verified
//
#include <hip/hip_runtime.h>

typedef __attribute__((ext_vector_type(2))) float v2f;
typedef __attribute__((ext_vector_type(8))) float v8f;

#define NB   2      // batch
#define CHN  4      // channels
#define SS   100    // image side
#define SP   112    // padded side (7 * 16)

// workspace layout (float offsets), total ~0.96 MB
#define OFF_WR    0                            // N*SP*SP     : Wr[n][qi][ik]
#define OFF_WCT   (NB*SP*SP)                   // N*SP*SP     : WcT[n][jk][qj]
#define OFF_MIN   (2*NB*SP*SP)                 // N*2*SP      : per-axis min dist^2
#define OFF_SUM   (2*NB*SP*SP + 2*NB*SP)       // N*2*SP      : per-axis weight sums
#define OFF_IMG   (2*NB*SP*SP + 4*NB*SP)       // N*CH*SP*SP  : zero-padded images
#define OFF_TMP   (OFF_IMG + NB*CHN*SP*SP)     // N*CH*SP*SP  : imgP @ WcT

// ---------------------------------------------------------------------------
// Step 0: per (n, axis, q): stabilization offset (min squared distance) and
// softmax partial denominator (sum of stabilized exp weights).
// axis 0 = rows (x coords), axis 1 = cols (y coords).
// ---------------------------------------------------------------------------
__global__ void dec_prep(const float* __restrict__ transforms, float* __restrict__ ws) {
    int t = blockIdx.x * blockDim.x + threadIdx.x;
    if (t >= NB * 2 * SP) return;
    int n    = t / (2 * SP);
    int rem  = t % (2 * SP);
    int axis = rem / SP;
    int q    = rem % SP;

    float t0 = transforms[n*4+0], t1 = transforms[n*4+1];
    float t2 = transforms[n*4+2], t3 = transforms[n*4+3];
    float scale = (axis == 0) ? (t1 - t0) : (t3 - t2);
    float off   = (axis == 0) ? (t0 * (float)SS) : (t2 * (float)SS);

    float m = 0.0f, sum = 1.0f;
    if (q < SS) {
        m = 3.4e38f;
        for (int k = 0; k < SS; ++k) {
            float d = (float)q - (scale * (float)k + off);
            m = fminf(m, d * d);
        }
        sum = 0.0f;
        for (int k = 0; k < SS; ++k) {
            float d = (float)q - (scale * (float)k + off);
            sum += __expf(m - d * d);
        }
    }
    ws[OFF_MIN + t] = m;
    ws[OFF_SUM + t] = sum;
}

// ---------------------------------------------------------------------------
// Step 1: fill zero-padded weight matrices (one block per padded row).
//   axis 0 row:  Wr [n][qi][ik] = exp(min_r(qi) - (qi - x(ik))^2)
//   axis 1 row:  WcT[n][jk][qj] = exp(min_c(qj) - (qj - y(jk))^2)
// ---------------------------------------------------------------------------
__global__ void dec_fill(const float* __restrict__ transforms, float* __restrict__ ws) {
    int b    = blockIdx.x;                 // N*2*SP rows
    int n    = b / (2 * SP);
    int rem  = b % (2 * SP);
    int axis = rem / SP;
    int row  = rem % SP;
    int col  = threadIdx.x;
    if (col >= SP) return;

    float t0 = transforms[n*4+0], t1 = transforms[n*4+1];
    float t2 = transforms[n*4+2], t3 = transforms[n*4+3];
    float scale = (axis == 0) ? (t1 - t0) : (t3 - t2);
    float off   = (axis == 0) ? (t0 * (float)SS) : (t2 * (float)SS);

    float val = 0.0f;
    if (row < SS && col < SS) {
        if (axis == 0) {   // Wr: row = qi, col = ik
            float m = ws[OFF_MIN + (n*2 + 0)*SP + row];
            float d = (float)row - (scale * (float)col + off);
            val = __expf(m - d * d);
        } else {           // WcT: row = jk, col = qj
            float m = ws[OFF_MIN + (n*2 + 1)*SP + col];
            float d = (float)col - (scale * (float)row + off);
            val = __expf(m - d * d);
        }
    }
    float* dst = (axis == 0) ? (ws + OFF_WR + n*SP*SP) : (ws + OFF_WCT + n*SP*SP);
    dst[row*SP + col] = val;
}

// ---------------------------------------------------------------------------
// Step 2: zero-pad images 100x100 -> 112x112 so the GEMMs are branch-free.
// One block per padded row of one (n,c) image.
// ---------------------------------------------------------------------------
__global__ void dec_pad(const float* __restrict__ images, float* __restrict__ ws) {
    int b   = blockIdx.x;                  // N*CH*SP rows
    int mat = b / SP;
    int row = b % SP;
    int col = threadIdx.x;
    if (col >= SP) return;
    float v = 0.0f;
    if (row < SS && col < SS) v = images[mat * SS * SS + row * SS + col];
    ws[OFF_IMG + mat * SP * SP + row * SP + col] = v;
}

// ---------------------------------------------------------------------------
// Wave-level 16x16 f32 WMMA tile: acc = A[tM*16.., :] @ B[:, tN*16..]
// A,B are SP x SP row-major, no bounds checks (padded). Full K unroll so the
// compiler can clause the loads and pipeline ahead of the WMMA chain.
// ---------------------------------------------------------------------------
__device__ __forceinline__ v8f wmma_tile_112(const float* __restrict__ A,
                                             const float* __restrict__ B,
                                             int tM, int tN, int lane) {
    int hs   = lane >> 4;       // half-wave select
    int l16  = lane & 15;
    int rowA = tM * 16 + l16;
    int colB = tN * 16 + l16;

    v8f c = {};
#pragma unroll
    for (int k0 = 0; k0 < SP; k0 += 4) {
        int ka = k0 + 2 * hs;   // frag VGPR0 holds K=ka, VGPR1 holds K=ka+1
        v2f a = *(const v2f*)(A + rowA * SP + ka);   // contiguous pair -> b64
        v2f b;
        b.x = B[ka * SP + colB];
        b.y = B[(ka + 1) * SP + colB];
        c = __builtin_amdgcn_wmma_f32_16x16x4_f32(false, a, false, b,
                                                  (short)0, c, false, false);
    }
    return c;
}

// ---------------------------------------------------------------------------
// GEMM 1: tmp[n,c] = imgP[n,c] @ WcT[n]   (112x112x112, one wave per tile)
// ---------------------------------------------------------------------------
__global__ void dec_gemm_img_wct(float* __restrict__ ws) {
    int bid  = blockIdx.x;            // N*CH*49
    int mat  = bid / 49;              // n*CH + c
    int tile = bid % 49;
    int tM = tile / 7, tN = tile % 7;
    int n  = mat / CHN;

    const float* A = ws + OFF_IMG + mat * SP * SP;
    const float* B = ws + OFF_WCT + n   * SP * SP;
    float*       D = ws + OFF_TMP + mat * SP * SP;

    int lane = threadIdx.x;
    v8f c = wmma_tile_112(A, B, tM, tN, lane);

    int hs = lane >> 4, l16 = lane & 15;
    int baseM = tM * 16 + hs * 8;
#pragma unroll
    for (int v = 0; v < 8; ++v)
        D[(baseM + v) * SP + tN * 16 + l16] = c[v];
}

// ---------------------------------------------------------------------------
// GEMM 2: out[n,c] = (Wr[n] @ tmp[n,c]) / (rowsum_r(qi) * rowsum_c(qj))
// ---------------------------------------------------------------------------
__global__ void dec_gemm_wr_tmp(const float* __restrict__ ws, float* __restrict__ out) {
    int bid  = blockIdx.x;            // N*CH*49
    int mat  = bid / 49;
    int tile = bid % 49;
    int tM = tile / 7, tN = tile % 7;
    int n  = mat / CHN;

    const float* A    = ws + OFF_WR  + n   * SP * SP;
    const float* B    = ws + OFF_TMP + mat * SP * SP;
    const float* rs_r = ws + OFF_SUM + (n*2 + 0) * SP;
    const float* rs_c = ws + OFF_SUM + (n*2 + 1) * SP;

    int lane = threadIdx.x;
    v8f c = wmma_tile_112(A, B, tM, tN, lane);

    int hs = lane >> 4, l16 = lane & 15;
    int qj = tN * 16 + l16;
    if (qj < SS) {
        float rc    = rs_c[qj];
        int   baseM = tM * 16 + hs * 8;
#pragma unroll
        for (int v = 0; v < 8; ++v) {
            int qi = baseM + v;
            if (qi < SS)
                out[mat * SS * SS + qi * SS + qj] = c[v] / (rs_r[qi] * rc);
        }
    }
}

// ---------------------------------------------------------------------------
extern "C" void kernel_launch(void* const* d_in, const int* in_sizes, int n_in,
                              void* d_out, int out_size, void* d_ws, size_t ws_size,
                              hipStream_t stream) {
    (void)in_sizes; (void)n_in; (void)out_size; (void)ws_size;
    const float* images     = (const float*)d_in[0];
    const float* transforms = (const float*)d_in[1];
    float*       out        = (float*)d_out;
    float*       ws         = (float*)d_ws;

    dec_prep<<<(NB*2*SP + 127) / 128, 128, 0, stream>>>(transforms, ws);
    dec_fill<<<NB*2*SP, 128, 0, stream>>>(transforms, ws);
    dec_pad <<<NB*CHN*SP, 128, 0, stream>>>(images, ws);
    dec_gemm_img_wct<<<NB*CHN*49, 32, 0, stream>>>(ws);
    dec_gemm_wr_tmp <<<NB*CHN*49, 32, 0, stream>>>(ws, out);
}